// SinkhornMatcher_45672682225973
// MI455X (gfx1250) — compile-verified
//
#include <hip/hip_runtime.h>
#include <math.h>

// CDNA5 wave32; full-fp32 WMMA path (V_WMMA_F32_16X16X4_F32).
typedef float v2f __attribute__((ext_vector_type(2)));
typedef float v8f __attribute__((ext_vector_type(8)));

#define NBATCH 16
#define KN_    128
#define DN     256

// workspace offsets (floats)
#define HA_OFF   0
#define HB_OFF   (NBATCH*KN_*DN)                 //  524288
#define DOT_OFF  (2*NBATCH*KN_*DN)               // 1048576
#define NA_OFF   (DOT_OFF + NBATCH*KN_*KN_)      // 1310720
#define NBN_OFF  (NA_OFF + NBATCH*KN_)           // 1312768
#define CT_OFF   (NBN_OFF + NBATCH*KN_)          // 1314816
// total ws: CT_OFF + 16*128*128 = 1576960 floats (~6.3 MB)

// output offsets (floats): T, cost, similarity, C (reference return order)
#define OUT_T_OFF    0
#define OUT_COST_OFF (NBATCH*KN_*KN_)            // 262144
#define OUT_SIM_OFF  (OUT_COST_OFF + NBATCH)     // 262160
#define OUT_C_OFF    (OUT_SIM_OFF + NBATCH)      // 262176

// ---------------------------------------------------------------------------
// Kernel 1: Ha[b,k,:] = nodes_a[b,k,:] @ W1[:256,:] + b1   (which==0)
//           Hb[b,m,:] = nodes_b[b,m,:] @ W1[256:,:]        (which==1)
// One wave per 16x16 output tile; fp32 WMMA, K=4, two accumulators.
// A-frag (16x4): lane L -> M = L&15, VGPR v -> K = v + 2*(L>>4)   [ISA 7.12.2]
// B-frag (4x16): lane L -> N = L&15, VGPR v -> K = v + 2*(L>>4)   [C/D-style striping]
// D (16x16):     VGPR v, lanes 0-15 -> M=v, lanes 16-31 -> M=v+8; N = L&15
// ---------------------------------------------------------------------------
__global__ __launch_bounds__(32)
void k_gemm_h(const float* __restrict__ nodes_a, const float* __restrict__ nodes_b,
              const float* __restrict__ W1, const float* __restrict__ b1,
              float* __restrict__ ws) {
  const int b     = blockIdx.x;
  const int kt    = blockIdx.y;
  const int z     = blockIdx.z;
  const int which = z >> 4;        // 0 = Ha, 1 = Hb
  const int nt    = z & 15;        // output column tile (16 tiles of 16 over DN)
  const int lane  = threadIdx.x;
  const int row   = lane & 15;
  const int half  = lane >> 4;

  const float* A = (which ? nodes_b : nodes_a) + (size_t)(b*KN_ + kt*16)*DN;
  const float* W = W1 + (size_t)which*DN*DN;   // rows [which*256, which*256+256)
  const int n    = nt*16 + row;

  const float* Arow = A + (size_t)row*DN + 2*half;
  const float* Wcol = W + (size_t)(2*half)*DN + n;

  v8f acc0 = {}; v8f acc1 = {};
#pragma unroll 4
  for (int e = 0; e < DN; e += 8) {
    v2f af0; af0.x = Arow[e];     af0.y = Arow[e+1];
    v2f bf0; bf0.x = Wcol[(size_t)e*DN];     bf0.y = Wcol[(size_t)(e+1)*DN];
    acc0 = __builtin_amdgcn_wmma_f32_16x16x4_f32(false, af0, false, bf0,
                                                 (short)0, acc0, false, false);
    v2f af1; af1.x = Arow[e+4];   af1.y = Arow[e+5];
    v2f bf1; bf1.x = Wcol[(size_t)(e+4)*DN]; bf1.y = Wcol[(size_t)(e+5)*DN];
    acc1 = __builtin_amdgcn_wmma_f32_16x16x4_f32(false, af1, false, bf1,
                                                 (short)0, acc1, false, false);
  }

  const float bias = which ? 0.0f : b1[n];     // fold b1 into Ha
  float* H = ws + (which ? HB_OFF : HA_OFF) + (size_t)(b*KN_ + kt*16)*DN + nt*16;
  const int col = row;
#pragma unroll
  for (int v = 0; v < 8; ++v)
    H[(size_t)(v + 8*half)*DN + col] = acc0[v] + acc1[v] + bias;
}

// ---------------------------------------------------------------------------
// Kernel 2: dot[b,k,m] = sum_e nodes_a[b,k,e] * nodes_b[b,m,e]  (A @ B^T)
// Since B = nodes_b^T, the B-fragment lanes read nodes_b row-major directly.
// ---------------------------------------------------------------------------
__global__ __launch_bounds__(32)
void k_gemm_dot(const float* __restrict__ nodes_a, const float* __restrict__ nodes_b,
                float* __restrict__ ws) {
  const int b = blockIdx.x, kt = blockIdx.y, mt = blockIdx.z;
  const int lane = threadIdx.x;
  const int row  = lane & 15;
  const int half = lane >> 4;

  const float* Ar = nodes_a + (size_t)(b*KN_ + kt*16 + row)*DN + 2*half;
  const float* Br = nodes_b + (size_t)(b*KN_ + mt*16 + row)*DN + 2*half;

  v8f acc0 = {}; v8f acc1 = {};
#pragma unroll 4
  for (int e = 0; e < DN; e += 8) {
    v2f af0; af0.x = Ar[e];   af0.y = Ar[e+1];
    v2f bf0; bf0.x = Br[e];   bf0.y = Br[e+1];
    acc0 = __builtin_amdgcn_wmma_f32_16x16x4_f32(false, af0, false, bf0,
                                                 (short)0, acc0, false, false);
    v2f af1; af1.x = Ar[e+4]; af1.y = Ar[e+5];
    v2f bf1; bf1.x = Br[e+4]; bf1.y = Br[e+5];
    acc1 = __builtin_amdgcn_wmma_f32_16x16x4_f32(false, af1, false, bf1,
                                                 (short)0, acc1, false, false);
  }

  float* Dst = ws + DOT_OFF + (size_t)b*KN_*KN_ + (size_t)(kt*16)*KN_ + mt*16;
  const int col = row;
#pragma unroll
  for (int v = 0; v < 8; ++v)
    Dst[(size_t)(v + 8*half)*KN_ + col] = acc0[v] + acc1[v];
}

// ---------------------------------------------------------------------------
// Kernel 3: row norms of nodes_a / nodes_b (wave per row, shfl reduction).
// ---------------------------------------------------------------------------
__global__ __launch_bounds__(128)
void k_norms(const float* __restrict__ nodes_a, const float* __restrict__ nodes_b,
             float* __restrict__ ws) {
  const int b     = blockIdx.x;
  const int which = blockIdx.y;
  const float* X  = (which ? nodes_b : nodes_a) + (size_t)b*KN_*DN;
  float* out      = ws + (which ? NBN_OFF : NA_OFF) + b*KN_;
  const int w    = threadIdx.x >> 5;
  const int lane = threadIdx.x & 31;
  for (int k = w; k < KN_; k += 4) {
    const float* rowp = X + (size_t)k*DN;
    float s = 0.f;
#pragma unroll
    for (int d = lane; d < DN; d += 32) { const float x = rowp[d]; s = fmaf(x, x, s); }
#pragma unroll
    for (int off = 16; off > 0; off >>= 1) s += __shfl_xor(s, off, 32);
    if (lane == 0) out[k] = sqrtf(s);
  }
}

// ---------------------------------------------------------------------------
// Kernel 4: scores + cosine + spatial -> C (and C^T for coalesced col pass).
// scores[k,m] = sum_d relu(Ha'[k,d] + Hb[m,d]) * W2[d] + b2   (Ha' has b1 folded)
// LDS tiles padded to pitch 260 floats: 16B-aligned float4 reads, banks spread.
// ---------------------------------------------------------------------------
__global__ __launch_bounds__(256)
void k_build_C(const float* __restrict__ pos_a, const float* __restrict__ pos_b,
               const float* __restrict__ W2, const float* __restrict__ b2,
               const float* __restrict__ log_temp,
               float* __restrict__ ws, float* __restrict__ out) {
  __shared__ float HaS[16][260];
  __shared__ float HbS[16][260];
  __shared__ float W2s[DN];

  const int b  = blockIdx.x;
  const int kt = blockIdx.y;
  const int tid = threadIdx.x;

  const float* Ha = ws + HA_OFF + (size_t)(b*KN_ + kt*16)*DN;
  for (int idx = tid; idx < 16*DN; idx += 256)
    HaS[idx >> 8][idx & 255] = Ha[(size_t)(idx >> 8)*DN + (idx & 255)];
  W2s[tid] = W2[tid];

  const float b2v = b2[0];
  float temp = expf(log_temp[0]);
  temp = fminf(fmaxf(temp, 0.01f), 0.5f);

  const int i = tid >> 4;          // k within tile
  const int j = tid & 15;          // m within tile
  const int k = kt*16 + i;
  const float nai = ws[NA_OFF + b*KN_ + k];
  const float pax = pos_a[(size_t)(b*KN_ + k)*2 + 0];
  const float pay = pos_a[(size_t)(b*KN_ + k)*2 + 1];

  for (int mt = 0; mt < 8; ++mt) {
    __syncthreads();
    const float* Hb = ws + HB_OFF + (size_t)(b*KN_ + mt*16)*DN;
    for (int idx = tid; idx < 16*DN; idx += 256)
      HbS[idx >> 8][idx & 255] = Hb[(size_t)(idx >> 8)*DN + (idx & 255)];
    __syncthreads();

    float s = 0.f;
#pragma unroll 8
    for (int d = 0; d < DN; d += 4) {
      const float4 ha = *(const float4*)&HaS[i][d];
      const float4 hb = *(const float4*)&HbS[j][d];
      const float4 w  = *(const float4*)&W2s[d];
      s = fmaf(fmaxf(ha.x + hb.x, 0.f), w.x, s);
      s = fmaf(fmaxf(ha.y + hb.y, 0.f), w.y, s);
      s = fmaf(fmaxf(ha.z + hb.z, 0.f), w.z, s);
      s = fmaf(fmaxf(ha.w + hb.w, 0.f), w.w, s);
    }

    const int m = mt*16 + j;
    const float score = s + b2v;
    const float dot   = ws[DOT_OFF + (size_t)b*KN_*KN_ + (size_t)k*KN_ + m];
    const float nbj   = ws[NBN_OFF + b*KN_ + m];
    const float cosv  = dot / fmaxf(nai*nbj, 1e-8f);
    const float pbx = pos_b[(size_t)(b*KN_ + m)*2 + 0];
    const float pby = pos_b[(size_t)(b*KN_ + m)*2 + 1];
    const float dx = pax - pbx, dy = pay - pby;
    const float spatial = sqrtf(dx*dx + dy*dy);
    const float Cv = -(score + cosv)/temp + 0.1f*spatial;

    out[OUT_C_OFF + (size_t)b*KN_*KN_ + (size_t)k*KN_ + m] = Cv;
    ws[CT_OFF     + (size_t)b*KN_*KN_ + (size_t)m*KN_ + k] = Cv;
  }
}

// ---------------------------------------------------------------------------
// Kernel 5: log-domain Sinkhorn (10 iters) + T, cost, similarity.
// One workgroup (8 wave32) per batch; log_u/log_v in LDS; wave-per-row lse.
// ---------------------------------------------------------------------------
__global__ __launch_bounds__(256)
void k_sinkhorn(float* __restrict__ ws, float* __restrict__ out) {
  __shared__ float lu[KN_], lv[KN_];
  __shared__ float naS[KN_], nbS[KN_];
  __shared__ float rc[8], rs[8];

  const int b   = blockIdx.x;
  const int tid = threadIdx.x;
  const int w    = tid >> 5;
  const int lane = tid & 31;

  const float* Cb  = out + OUT_C_OFF + (size_t)b*KN_*KN_;
  const float* Ctb = ws  + CT_OFF    + (size_t)b*KN_*KN_;
  const float* Db  = ws  + DOT_OFF   + (size_t)b*KN_*KN_;

  if (tid < KN_) {
    lv[tid]  = 0.f;
    naS[tid] = ws[NA_OFF  + b*KN_ + tid];
    nbS[tid] = ws[NBN_OFF + b*KN_ + tid];
  }
  const float LOGMU = -logf((float)KN_);
  __syncthreads();

  for (int it = 0; it < 10; ++it) {
    // log_u[k] = log_mu - lse_m( C[k,m] + log_v[m] )
    for (int k = w; k < KN_; k += 8) {
      const float* rowp = Cb + (size_t)k*KN_;
      float x0 = rowp[lane]      + lv[lane];
      float x1 = rowp[lane + 32] + lv[lane + 32];
      float x2 = rowp[lane + 64] + lv[lane + 64];
      float x3 = rowp[lane + 96] + lv[lane + 96];
      float mx = fmaxf(fmaxf(x0, x1), fmaxf(x2, x3));
#pragma unroll
      for (int off = 16; off > 0; off >>= 1) mx = fmaxf(mx, __shfl_xor(mx, off, 32));
      float s = expf(x0 - mx) + expf(x1 - mx) + expf(x2 - mx) + expf(x3 - mx);
#pragma unroll
      for (int off = 16; off > 0; off >>= 1) s += __shfl_xor(s, off, 32);
      if (lane == 0) lu[k] = LOGMU - (mx + logf(s));
    }
    __syncthreads();
    // log_v[m] = log_nu - lse_k( C[k,m] + log_u[k] )  (coalesced via C^T)
    for (int m = w; m < KN_; m += 8) {
      const float* colp = Ctb + (size_t)m*KN_;
      float x0 = colp[lane]      + lu[lane];
      float x1 = colp[lane + 32] + lu[lane + 32];
      float x2 = colp[lane + 64] + lu[lane + 64];
      float x3 = colp[lane + 96] + lu[lane + 96];
      float mx = fmaxf(fmaxf(x0, x1), fmaxf(x2, x3));
#pragma unroll
      for (int off = 16; off > 0; off >>= 1) mx = fmaxf(mx, __shfl_xor(mx, off, 32));
      float s = expf(x0 - mx) + expf(x1 - mx) + expf(x2 - mx) + expf(x3 - mx);
#pragma unroll
      for (int off = 16; off > 0; off >>= 1) s += __shfl_xor(s, off, 32);
      if (lane == 0) lv[m] = LOGMU - (mx + logf(s));
    }
    __syncthreads();
  }

  float cacc = 0.f, sacc = 0.f;
  for (int idx = tid; idx < KN_*KN_; idx += 256) {
    const int k = idx >> 7, m = idx & 127;
    const float Cv = Cb[idx];
    const float Tv = expf(Cv + lu[k] + lv[m]);
    out[OUT_T_OFF + (size_t)b*KN_*KN_ + idx] = Tv;
    const float cosv = Db[idx] / fmaxf(naS[k]*nbS[m], 1e-8f);
    cacc = fmaf(Tv, Cv, cacc);
    sacc = fmaf(Tv, cosv, sacc);
  }
#pragma unroll
  for (int off = 16; off > 0; off >>= 1) {
    cacc += __shfl_xor(cacc, off, 32);
    sacc += __shfl_xor(sacc, off, 32);
  }
  if (lane == 0) { rc[w] = cacc; rs[w] = sacc; }
  __syncthreads();
  if (tid == 0) {
    float c = 0.f, s = 0.f;
#pragma unroll
    for (int q = 0; q < 8; ++q) { c += rc[q]; s += rs[q]; }
    out[OUT_COST_OFF + b] = c;
    out[OUT_SIM_OFF  + b] = s;
  }
}

// ---------------------------------------------------------------------------
extern "C" void kernel_launch(void* const* d_in, const int* in_sizes, int n_in,
                              void* d_out, int out_size, void* d_ws, size_t ws_size,
                              hipStream_t stream) {
  (void)in_sizes; (void)n_in; (void)out_size; (void)ws_size;
  const float* nodes_a  = (const float*)d_in[0];
  const float* nodes_b  = (const float*)d_in[1];
  const float* pos_a    = (const float*)d_in[2];
  const float* pos_b    = (const float*)d_in[3];
  const float* W1       = (const float*)d_in[4];
  const float* b1       = (const float*)d_in[5];
  const float* W2       = (const float*)d_in[6];
  const float* b2       = (const float*)d_in[7];
  const float* log_temp = (const float*)d_in[8];
  float* ws  = (float*)d_ws;
  float* out = (float*)d_out;

  // Ha/Hb GEMMs: z = which*16 + ntile
  k_gemm_h  <<<dim3(NBATCH, 8, 32), 32, 0, stream>>>(nodes_a, nodes_b, W1, b1, ws);
  // dot = A @ B^T
  k_gemm_dot<<<dim3(NBATCH, 8, 8),  32, 0, stream>>>(nodes_a, nodes_b, ws);
  // row norms
  k_norms   <<<dim3(NBATCH, 2),    128, 0, stream>>>(nodes_a, nodes_b, ws);
  // cost matrix C (+ C^T)
  k_build_C <<<dim3(NBATCH, 8),    256, 0, stream>>>(pos_a, pos_b, W2, b2, log_temp, ws, out);
  // Sinkhorn + outputs
  k_sinkhorn<<<NBATCH,             256, 0, stream>>>(ws, out);
}